// GNNGUARD_51505247814280
// MI455X (gfx1250) — compile-verified
//
#include <hip/hip_runtime.h>
#include <hip/hip_bf16.h>

typedef __attribute__((ext_vector_type(2))) float v2f;
typedef __attribute__((ext_vector_type(8))) float v8f;

#define D_FEAT 48
#define THRESH 0.1f
#define EPS_N  1e-8f

// ---------------------------------------------------------------------------
// Kernel 1: per-node inverse norm + zero the segment-sum accumulator.
// 48 floats = 12 float4 loads per node; 100K nodes, trivially memory bound.
// ---------------------------------------------------------------------------
__global__ void gnn_node_kernel(const float* __restrict__ x,
                                float* __restrict__ inv_norm,
                                float* __restrict__ row_sum,
                                int n_nodes)
{
    int i = blockIdx.x * blockDim.x + threadIdx.x;
    if (i >= n_nodes) return;
    const float4* p = (const float4*)(x + (size_t)i * D_FEAT);
    float s = 0.f;
#pragma unroll
    for (int j = 0; j < D_FEAT / 4; ++j) {
        float4 v = p[j];
        s = fmaf(v.x, v.x, s);
        s = fmaf(v.y, v.y, s);
        s = fmaf(v.z, v.z, s);
        s = fmaf(v.w, v.w, s);
    }
    float nrm = sqrtf(s);
    inv_norm[i] = 1.0f / fmaxf(nrm, EPS_N);
    row_sum[i]  = 0.0f;
}

// ---------------------------------------------------------------------------
// Kernel 2: pass edge_index through to the (float) output buffer.
// ---------------------------------------------------------------------------
__global__ void gnn_idx_kernel(const int* __restrict__ ei,
                               float* __restrict__ out,
                               int n)
{
    int i = blockIdx.x * blockDim.x + threadIdx.x;
    if (i < n) out[i] = (float)ei[i];
}

// ---------------------------------------------------------------------------
// Kernel 3: per-edge cosine similarity via V_WMMA_F32_16X16X4_F32.
// One wave32 handles 16 edges. A = 16 edges x 4 feats of endpoint-A rows,
// B = 4 feats x 16 edges of endpoint-B rows; 12 WMMAs cover K=48. The
// diagonal of the 16x16 f32 accumulator is the per-edge dot product.
//
// A-operand layout (32-bit 16x4): VGPR0 = K0 (lanes 0-15) / K2 (lanes 16-31),
// VGPR1 = K1 / K3. B mirrors it with N striped across lanes. So lane l
// (edge m = l&15) loads features [k + 2*(l>>4)] and [.. + 1] of its row.
//
// Diagonal (m,m) of C: m<8 -> acc[m] @ lane m; m>=8 -> acc[m-8] @ lane m+16.
// ---------------------------------------------------------------------------
__global__ void gnn_edge_wmma_kernel(const float* __restrict__ x,
                                     const int* __restrict__ row,
                                     const int* __restrict__ col,
                                     const float* __restrict__ inv_norm,
                                     float* __restrict__ att,
                                     float* __restrict__ row_sum,
                                     int n_edges)
{
    const int lane  = threadIdx.x & 31;
    const int wave  = (blockIdx.x * blockDim.x + threadIdx.x) >> 5;
    const int ebase = wave * 16;
    if (ebase >= n_edges) return;

    if (ebase + 16 <= n_edges) {
        // -------- full tile: WMMA path (EXEC is all ones) --------
        const int m = lane & 15;          // edge-in-tile for this lane
        const int e = ebase + m;
        const int r = row[e];
        const int c = col[e];
        const float* ap = x + (size_t)r * D_FEAT;
        const float* bp = x + (size_t)c * D_FEAT;
        const int koff = (lane >> 4) << 1;   // 0 for lanes 0-15, 2 for 16-31

        v8f acc = {};
#pragma unroll
        for (int j = 0; j < D_FEAT / 4; ++j) {
            const int k = j * 4 + koff;
            v2f a, b;
            a.x = ap[k];
            a.y = ap[k + 1];
            b.x = bp[k];
            b.y = bp[k + 1];
            acc = __builtin_amdgcn_wmma_f32_16x16x4_f32(
                /*neg_a=*/false, a, /*neg_b=*/false, b,
                /*c_mod=*/(short)0, acc, /*reuse_a=*/false, /*reuse_b=*/false);
        }

        // extract diagonal: 16 shuffles, every lane ends with dot for edge m
        float dot = 0.f;
#pragma unroll
        for (int rr = 0; rr < 8; ++rr) {
            float vlo = __shfl(acc[rr], rr, 32);        // diag (rr, rr)
            float vhi = __shfl(acc[rr], rr + 24, 32);   // diag (rr+8, rr+8)
            dot = (m == rr)     ? vlo : dot;
            dot = (m == rr + 8) ? vhi : dot;
        }

        if (lane < 16) {
            float s = dot * inv_norm[r] * inv_norm[c];
            s = (s < THRESH) ? 0.0f : s;
            att[e] = s;
            if (s != 0.0f) atomicAdd(&row_sum[r], s);
        }
    } else {
        // -------- ragged tail: scalar per-lane dot (wave-uniform branch) ----
        if (lane < n_edges - ebase) {
            const int e = ebase + lane;
            const int r = row[e];
            const int c = col[e];
            const float* ap = x + (size_t)r * D_FEAT;
            const float* bp = x + (size_t)c * D_FEAT;
            float dot = 0.f;
#pragma unroll
            for (int k = 0; k < D_FEAT; ++k) dot = fmaf(ap[k], bp[k], dot);
            float s = dot * inv_norm[r] * inv_norm[c];
            s = (s < THRESH) ? 0.0f : s;
            att[e] = s;
            if (s != 0.0f) atomicAdd(&row_sum[r], s);
        }
    }
}

// ---------------------------------------------------------------------------
// Kernel 4: row-wise L1 normalization (zero rows unchanged) + exp.
// ---------------------------------------------------------------------------
__global__ void gnn_finalize_kernel(const int* __restrict__ row,
                                    const float* __restrict__ att,
                                    const float* __restrict__ row_sum,
                                    float* __restrict__ out,
                                    int n_edges)
{
    int e = blockIdx.x * blockDim.x + threadIdx.x;
    if (e >= n_edges) return;
    float rs    = row_sum[row[e]];
    float denom = (rs == 0.0f) ? 1.0f : rs;
    out[e] = expf(att[e] / denom);
}

// ---------------------------------------------------------------------------
extern "C" void kernel_launch(void* const* d_in, const int* in_sizes, int n_in,
                              void* d_out, int out_size, void* d_ws, size_t ws_size,
                              hipStream_t stream) {
    const float* x  = (const float*)d_in[0];
    const int*   ei = (const int*)d_in[1];
    const int N = in_sizes[0] / D_FEAT;   // 100000
    const int E = in_sizes[1] / 2;        // 1600000
    const int* row = ei;
    const int* col = ei + E;

    // workspace layout: att[E] | row_sum[N] | inv_norm[N]
    float* att      = (float*)d_ws;
    float* row_sum  = att + E;
    float* inv_norm = row_sum + N;

    float* out      = (float*)d_out;      // [0,2E): edge_index, [2E,3E): scores
    float* out_att  = out + 2 * (size_t)E;

    // 1) per-node inv-norms + zero row sums
    gnn_node_kernel<<<(N + 255) / 256, 256, 0, stream>>>(x, inv_norm, row_sum, N);

    // 2) edge_index passthrough
    gnn_idx_kernel<<<(2 * E + 255) / 256, 256, 0, stream>>>(ei, out, 2 * E);

    // 3) WMMA cosine-similarity + threshold + segment-sum (16 edges / wave,
    //    256-thread block = 8 waves = 128 edges; 1.6M/128 = 12500 blocks exact)
    gnn_edge_wmma_kernel<<<(E + 127) / 128, 256, 0, stream>>>(
        x, row, col, inv_norm, att, row_sum, E);

    // 4) normalize + exp
    gnn_finalize_kernel<<<(E + 255) / 256, 256, 0, stream>>>(
        row, att, row_sum, out_att, E);
}